// MultiOrder_58299886076400
// MI455X (gfx1250) — compile-verified
//
#include <hip/hip_runtime.h>
#include <math.h>

typedef __attribute__((ext_vector_type(2))) float v2f;
typedef __attribute__((ext_vector_type(8))) float v8f;

#define KSIZE 21
#define HALO  10   // (KSIZE-1)/2

__device__ __forceinline__ int reflect_idx(int i, int M) {
    // jnp.pad(..., mode="reflect"): -1 -> 1, M -> M-2
    i = (i < 0) ? -i : i;
    i = (i >= M) ? (2 * M - 2 - i) : i;
    return i;
}

// ---------------------------------------------------------------------------
// Stage 1: 21-tap Gaussian blur as an im2col GEMM on the WMMA pipe.
// One wave32 produces 256 consecutive smoothed outputs as a 16x16 tile:
//   Out[r,c] = smoothed[base + 16r + c] = sum_{j=0..35} A[r,j] * B[j,c]
//   A[r,j] = x_reflect[base + 16r + j - 10],  B[j,c] = k[j-c] (0 outside [0,20])
// K = 36 -> 9 chained V_WMMA_F32_16X16X4_F32 (accumulator D feeds C).
// ISA f32 layouts (§7.12.2):
//   A 16x4 : lane L(0..15) holds K=0,1 in v0,v1 ; lane L+16 holds K=2,3
//   B 4x16 : lane L(0..15) holds rows K=0,1 (col=L) ; lane L+16 rows K=2,3
//   C/D    : VGPR v = row v (lanes 0..15) / row v+8 (lanes 16..31), col = L&15
// ---------------------------------------------------------------------------
__global__ void blur_wmma_kernel(const float* __restrict__ x,
                                 const float* __restrict__ log_blur_size,
                                 float* __restrict__ smoothed, int M) {
    const int lane   = threadIdx.x & 31;
    const int wave   = threadIdx.x >> 5;
    const int tile   = blockIdx.x * (blockDim.x >> 5) + wave;
    const int ntiles = M >> 8;
    if (tile >= ntiles) return;          // wave-uniform: EXEC stays all-ones
    const int base = tile << 8;

    // Normalized Gaussian kernel coefficients (recomputed per lane; ~40 expf).
    const float sigma   = expf(log_blur_size[0]);
    const float inv_s2  = 1.0f / (sigma * sigma);
    float ksum = 0.0f;
#pragma unroll
    for (int t = 0; t < KSIZE; ++t) {
        float tt = (float)t - (float)HALO;
        ksum += expf(-0.5f * tt * tt * inv_s2);
    }
    const float inv_ksum = 1.0f / ksum;

    auto kco = [&](int t) -> float {
        float tt = (float)t - (float)HALO;
        float v  = expf(-0.5f * tt * tt * inv_s2) * inv_ksum;
        return ((unsigned)t <= 20u) ? v : 0.0f;   // banded: zero outside taps
    };

    const int mrow = lane & 15;        // A row (and B column) index
    const int krow = (lane >> 4) << 1; // 0 for lanes 0-15, 2 for lanes 16-31

    v8f acc = {0.f, 0.f, 0.f, 0.f, 0.f, 0.f, 0.f, 0.f};

#pragma unroll
    for (int kk = 0; kk < 9; ++kk) {
        const int j0 = 4 * kk + krow;
        v2f a, b;
        const int g = base + mrow * 16 + j0 - HALO;
        a.x = x[reflect_idx(g,     M)];
        a.y = x[reflect_idx(g + 1, M)];
        b.x = kco(j0 - mrow);          // B[j0,   col]
        b.y = kco(j0 + 1 - mrow);      // B[j0+1, col]
        acc = __builtin_amdgcn_wmma_f32_16x16x4_f32(
            /*neg_a=*/false, a, /*neg_b=*/false, b,
            /*c_mod=*/(short)0, acc, /*reuse_a=*/false, /*reuse_b=*/false);
    }

    const int hi  = lane >> 4;
    const int col = lane & 15;
#pragma unroll
    for (int v = 0; v < 8; ++v) {
        const int row = v + 8 * hi;
        smoothed[base + row * 16 + col] = acc[v];
    }
}

// ---------------------------------------------------------------------------
// Stage 2: nearest-pixel binning, inverted as a per-bin gather (deterministic,
// no atomics). Both grids are linspaces, so bin n owns a contiguous native
// window (~9 samples). Scan window + guard band; membership uses the faithful
// argmin tie rule against actual wl_data neighbors (first index wins ties).
// ---------------------------------------------------------------------------
__global__ void bin_kernel(const float* __restrict__ wl_data,
                           const float* __restrict__ wl_native,
                           const float* __restrict__ smoothed,
                           const float* __restrict__ scalar_const,
                           const float* __restrict__ v_z,
                           float* __restrict__ out, int N, int M) {
    const int n = blockIdx.x * blockDim.x + threadIdx.x;
    if (n >= N) return;

    const float C  = 299792.458f;
    const float vz = v_z[0];
    const float s  = sqrtf((C + vz) / (C - vz));   // rv_shift

    const float wl0 = wl_data[0];
    const float wlN = wl_data[N - 1];
    const float d   = (wlN - wl0) / (float)(N - 1);

    const float a0 = wl_native[0];
    const float b  = (wl_native[M - 1] - a0) / (float)(M - 1);

    const float wc    = wl_data[n];
    const float wlft  = (n > 0)     ? wl_data[n - 1] : wc;
    const float wrgt  = (n < N - 1) ? wl_data[n + 1] : wc;

    // Predicted native-index window for this bin + guard band.
    const float mc   = (wc / s - a0) / b;
    const float half = 0.5f * d / (b * s) + 4.0f;
    int mlo = (int)floorf(mc - half); if (mlo < 0)     mlo = 0;
    int mhi = (int)ceilf (mc + half); if (mhi > M - 1) mhi = M - 1;

    float num = 0.0f, den = 0.0f;
    for (int m = mlo; m <= mhi; ++m) {
        const float ws = wl_native[m] * s;
        if (!(ws > wl0 && ws < wlN)) continue;      // trim mask weight
        const float dc   = wc - ws;
        const float dist = dc * dc;
        bool left_ok = (n == 0);
        if (!left_ok) { const float dl = wlft - ws; left_ok = dist <  dl * dl; }
        bool right_ok = (n == N - 1);
        if (!right_ok){ const float dr = wrgt - ws; right_ok = dist <= dr * dr; }
        if (left_ok && right_ok) { num += smoothed[m]; den += 1.0f; }
    }
    out[n] = (den > 0.0f) ? (num / den) * scalar_const[0] : 0.0f;
}

extern "C" void kernel_launch(void* const* d_in, const int* in_sizes, int n_in,
                              void* d_out, int out_size, void* d_ws, size_t ws_size,
                              hipStream_t stream) {
    const float* wl_data      = (const float*)d_in[0];
    const float* wl_native    = (const float*)d_in[1];
    const float* flux_native  = (const float*)d_in[2];
    const float* scalar_const = (const float*)d_in[3];
    const float* v_z          = (const float*)d_in[4];
    const float* log_blur     = (const float*)d_in[5];

    const int N = in_sizes[0];   // 4096 detector pixels
    const int M = in_sizes[2];   // 65536 native samples (multiple of 256)

    float* smoothed = (float*)d_ws;  // M floats of scratch

    // Stage 1: one wave per 256-output tile, 4 waves (128 threads) per block.
    const int ntiles = M >> 8;
    const int wpb    = 4;
    const int blocks = (ntiles + wpb - 1) / wpb;
    blur_wmma_kernel<<<blocks, wpb * 32, 0, stream>>>(flux_native, log_blur,
                                                      smoothed, M);

    // Stage 2: one thread per output bin.
    const int tb = 256;
    bin_kernel<<<(N + tb - 1) / tb, tb, 0, stream>>>(wl_data, wl_native,
                                                     smoothed, scalar_const,
                                                     v_z, (float*)d_out, N, M);
}